// CenterLoss_84344567758854
// MI455X (gfx1250) — compile-verified
//
#include <hip/hip_runtime.h>

typedef __attribute__((ext_vector_type(2))) float v2f;
typedef __attribute__((ext_vector_type(8))) float v8f;

#define BATCH     4096
#define DIM       512
#define NCENTERS  10000
#define CLAMP_MIN 1e-12f
#define CLAMP_MAX 1e12f

// Kernel 1: one wave (32 lanes) computes 16 rows' squared distances to their
// label centers. dot(x_row, c_row) comes from the diagonal of a 16x16 cross-
// Gram tile accumulated with v_wmma_f32_16x16x4_f32; ||x||^2 and ||c||^2 are
// per-lane VALU partial sums folded with a shfl_xor(16).
__global__ __launch_bounds__(256) void CenterLoss_rows_kernel(
    const float* __restrict__ x, const int* __restrict__ labels,
    const float* __restrict__ centers, float* __restrict__ row_d) {
  const int lane = threadIdx.x & 31;
  const int waveInBlock = threadIdx.x >> 5;
  const int tile = blockIdx.x * 8 + waveInBlock;   // 16 rows per wave
  const int m = lane & 15;                         // row within tile
  const int row = tile * 16 + m;
  const int ksel = (lane >> 4) * 2;                // half-wave K-pair select

  const int label = labels[row];
  const float* __restrict__ xr = x + (size_t)row * DIM;
  const float* __restrict__ cr = centers + (size_t)label * DIM;

  v8f acc = {};                 // 16x16 f32 Gram tile: acc[m][n] = x_m . c_n
  float sxx = 0.0f, scc = 0.0f;

#pragma unroll 4
  for (int k0 = 0; k0 < DIM; k0 += 4) {
    // A (16x4 f32): lane m holds x_m[k0+ksel], x_m[k0+ksel+1] in V0,V1.
    // B (4x16 f32): lane n holds c_n[k0+ksel], c_n[k0+ksel+1] in V0,V1.
    // Identical per-lane indexing -> one float2 load each.
    v2f xa = *(const v2f*)(xr + k0 + ksel);
    v2f ca = *(const v2f*)(cr + k0 + ksel);
    sxx += xa.x * xa.x + xa.y * xa.y;
    scc += ca.x * ca.x + ca.y * ca.y;
    acc = __builtin_amdgcn_wmma_f32_16x16x4_f32(
        /*neg_a=*/false, xa, /*neg_b=*/false, ca,
        /*c_mod=*/(short)0, acc, /*reuse_a=*/false, /*reuse_b=*/false);
  }

  // Each row's |x|^2 / |c|^2 partials live in lanes m and m+16.
  sxx += __shfl_xor(sxx, 16, 32);
  scc += __shfl_xor(scc, 16, 32);

  // Diagonal of the Gram tile (C/D layout: VGPR r, lanes 0-15 -> M=r, N=lane;
  // lanes 16-31 -> M=8+r, N=lane-16). So diag m<8 is acc[m] @ lane m and
  // diag m>=8 is acc[m-8] @ lane 24+(m-8).
  float dot = 0.0f;
#pragma unroll
  for (int r = 0; r < 8; ++r) {
    float d0 = __shfl(acc[r], r, 32);        // diag element r
    float d1 = __shfl(acc[r], 24 + r, 32);   // diag element 8 + r
    dot = (lane == r) ? d0 : dot;
    dot = (lane == 8 + r) ? d1 : dot;
  }

  if (lane < 16) {
    float d = sxx + scc - 2.0f * dot;        // matches reference formula
    d = fminf(fmaxf(d, CLAMP_MIN), CLAMP_MAX);
    row_d[row] = d;
  }
}

// Kernel 2: deterministic single-block reduction of the 4096 clipped row
// distances, plus the closed-form B*(C-1)*1e-12 contribution from the
// masked-then-clamped zero entries.
__global__ __launch_bounds__(256) void CenterLoss_reduce_kernel(
    const float* __restrict__ row_d, float* __restrict__ out) {
  __shared__ float smem[256];
  float s = 0.0f;
  for (int i = threadIdx.x; i < BATCH; i += 256) s += row_d[i];
  smem[threadIdx.x] = s;
  __syncthreads();
#pragma unroll
  for (int stride = 128; stride > 0; stride >>= 1) {
    if (threadIdx.x < stride) smem[threadIdx.x] += smem[threadIdx.x + stride];
    __syncthreads();
  }
  if (threadIdx.x == 0) {
    double total = (double)smem[0] +
                   (double)BATCH * (double)(NCENTERS - 1) * 1e-12;
    out[0] = (float)(total / (double)BATCH);
  }
}

extern "C" void kernel_launch(void* const* d_in, const int* in_sizes, int n_in,
                              void* d_out, int out_size, void* d_ws, size_t ws_size,
                              hipStream_t stream) {
  const float* x       = (const float*)d_in[0];
  const int*   labels  = (const int*)d_in[1];
  const float* centers = (const float*)d_in[2];
  float* row_d = (float*)d_ws;   // 4096 floats of scratch

  // 32 blocks * 8 waves * 16 rows = 4096 rows
  CenterLoss_rows_kernel<<<BATCH / (16 * 8), 256, 0, stream>>>(
      x, labels, centers, row_d);
  CenterLoss_reduce_kernel<<<1, 256, 0, stream>>>(row_d, (float*)d_out);
}